// GAT_7988639171254
// MI455X (gfx1250) — compile-verified
//
#include <hip/hip_runtime.h>

#define HIDDEN   128
#define HEADS    4
#define HEAD_DIM 32
#define NEG_SLOPE 0.2f

typedef __attribute__((ext_vector_type(2))) float v2f;
typedef __attribute__((ext_vector_type(8))) float v8f;

// ---------------------------------------------------------------------------
// Fill kernel (used to zero accumulators / init max-keys each launch)
// ---------------------------------------------------------------------------
__global__ void gat_fill_u32(unsigned* __restrict__ p, unsigned v, long long n) {
    long long i = (long long)blockIdx.x * blockDim.x + threadIdx.x;
    if (i < n) p[i] = v;
}

// ---------------------------------------------------------------------------
// GEMM: H[N,128] = X[N,128] @ W[128,128] using V_WMMA_F32_16X16X4_F32.
// Block = 256 threads = 8 waves. Each wave computes one 16x16 tile:
// rows = blockIdx.x*16 .. +15, cols = wave*16 .. +15 (8 waves * 16 = 128).
// fp32 A 16x4 fragment: lane = m + 16*half, v0 holds K=2*half, v1 K=2*half+1.
// fp32 B 4x16 fragment: lane = n + 16*half, v0 holds K=2*half, v1 K=2*half+1.
// fp32 C/D 16x16: lane = n + 16*half, VGPR v holds row m = v + 8*half.
// ---------------------------------------------------------------------------
__global__ __launch_bounds__(256) void gat_gemm_wmma(
    const float* __restrict__ X, const float* __restrict__ W,
    float* __restrict__ H, int N)
{
    const int wave = threadIdx.x >> 5;
    const int lane = threadIdx.x & 31;
    const int half = lane >> 4;
    const int l16  = lane & 15;
    const int row  = blockIdx.x * 16;
    const int col  = wave * 16;

    int r = row + l16;
    if (r >= N) r = N - 1;                 // clamp (no EXEC divergence at WMMA)
    const float* __restrict__ xrow = X + (long long)r * HIDDEN;

    v8f c = {};
    #pragma unroll
    for (int k0 = 0; k0 < HIDDEN; k0 += 4) {
        const int k = k0 + 2 * half;
        v2f a, b;
        a.x = xrow[k];
        a.y = xrow[k + 1];
        b.x = W[(long long)k       * HIDDEN + col + l16];
        b.y = W[(long long)(k + 1) * HIDDEN + col + l16];
        c = __builtin_amdgcn_wmma_f32_16x16x4_f32(
                false, a, false, b, (short)0, c, false, false);
    }
    #pragma unroll
    for (int v = 0; v < 8; ++v) {
        const int rr = row + v + 8 * half;
        if (rr < N) H[(long long)rr * HIDDEN + col + l16] = c[v];
    }
}

// ---------------------------------------------------------------------------
// Per-(node,head) attention coefficients: a_src/a_dst = <h[n,h,:], att[h,:]>
// ---------------------------------------------------------------------------
__global__ void gat_attn_coeff(const float* __restrict__ H,
                               const float* __restrict__ att_s,
                               const float* __restrict__ att_d,
                               float* __restrict__ a_s,
                               float* __restrict__ a_d, int N)
{
    const int t = blockIdx.x * blockDim.x + threadIdx.x;
    if (t >= N * HEADS) return;
    const int h = t & (HEADS - 1);
    const int n = t >> 2;
    const float* __restrict__ hp = H + (long long)n * HIDDEN + h * HEAD_DIM;
    const float* __restrict__ as = att_s + h * HEAD_DIM;
    const float* __restrict__ ad = att_d + h * HEAD_DIM;
    float ss = 0.f, sd = 0.f;
    #pragma unroll
    for (int c = 0; c < HEAD_DIM; ++c) {
        const float v = hp[c];
        ss += v * as[c];
        sd += v * ad[c];
    }
    a_s[t] = ss;
    a_d[t] = sd;
}

// order-preserving float <-> uint encoding for atomicMax
__device__ __forceinline__ unsigned gat_fenc(float f) {
    unsigned b = __float_as_uint(f);
    return (b & 0x80000000u) ? ~b : (b | 0x80000000u);
}
__device__ __forceinline__ float gat_fdec(unsigned e) {
    unsigned b = (e & 0x80000000u) ? (e ^ 0x80000000u) : ~e;
    return __uint_as_float(b);
}

__device__ __forceinline__ void gat_edge_sd(long long t, const int* __restrict__ ei,
                                            int E, int& s, int& d) {
    if (t < E) { s = ei[t]; d = ei[E + t]; }
    else       { s = (int)(t - E); d = s; }   // self loop
}

// ---------------------------------------------------------------------------
// Pass A: segment max over dst of leaky_relu(a_src[s] + a_dst[d])
// ---------------------------------------------------------------------------
__global__ void gat_edge_max(const float* __restrict__ a_s,
                             const float* __restrict__ a_d,
                             const int* __restrict__ ei, int E, long long EE,
                             unsigned* __restrict__ mk)
{
    const long long t = (long long)blockIdx.x * blockDim.x + threadIdx.x;
    if (t >= EE * HEADS) return;
    const long long e = t >> 2;
    const int h = (int)(t & (HEADS - 1));
    int s, d;
    gat_edge_sd(e, ei, E, s, d);
    float v = a_s[(long long)s * HEADS + h] + a_d[(long long)d * HEADS + h];
    v = (v > 0.f) ? v : NEG_SLOPE * v;
    atomicMax(&mk[(long long)d * HEADS + h], gat_fenc(v));
}

// ---------------------------------------------------------------------------
// Pass B: ex = exp(e - max[dst]); store ex; denom[dst] += ex
// ---------------------------------------------------------------------------
__global__ void gat_edge_exp(const float* __restrict__ a_s,
                             const float* __restrict__ a_d,
                             const int* __restrict__ ei, int E, long long EE,
                             const unsigned* __restrict__ mk,
                             float* __restrict__ exb,
                             float* __restrict__ den)
{
    const long long t = (long long)blockIdx.x * blockDim.x + threadIdx.x;
    if (t >= EE * HEADS) return;
    const long long e = t >> 2;
    const int h = (int)(t & (HEADS - 1));
    int s, d;
    gat_edge_sd(e, ei, E, s, d);
    float v = a_s[(long long)s * HEADS + h] + a_d[(long long)d * HEADS + h];
    v = (v > 0.f) ? v : NEG_SLOPE * v;
    const float mx = gat_fdec(mk[(long long)d * HEADS + h]);
    const float ex = __expf(v - mx);
    exb[t] = ex;
    atomicAdd(&den[(long long)d * HEADS + h], ex);
}

// ---------------------------------------------------------------------------
// Pass C: out[dst] += h[src] * alpha.  One wave (32 lanes) per edge; lane
// handles channel `lane` of each of the 4 heads (4 atomic f32 adds/lane).
// ---------------------------------------------------------------------------
__global__ __launch_bounds__(256) void gat_edge_aggr(
    const float* __restrict__ H, const float* __restrict__ exb,
    const float* __restrict__ den, const int* __restrict__ ei,
    int E, long long EE, float* __restrict__ acc)
{
    const int wave = threadIdx.x >> 5;
    const int lane = threadIdx.x & 31;
    const long long t = (long long)blockIdx.x * 8 + wave;
    if (t >= EE) return;
    int s, d;
    gat_edge_sd(t, ei, E, s, d);
    const float4 ex4 = *(const float4*)(exb + t * HEADS);
    const float4 dn4 = *(const float4*)(den + (long long)d * HEADS);
    const float al[4] = { ex4.x / (dn4.x + 1e-16f), ex4.y / (dn4.y + 1e-16f),
                          ex4.z / (dn4.z + 1e-16f), ex4.w / (dn4.w + 1e-16f) };
    const float* __restrict__ hs = H + (long long)s * HIDDEN + lane;
    float* __restrict__ ao = acc + (long long)d * HIDDEN + lane;
    #pragma unroll
    for (int j = 0; j < HEADS; ++j)
        atomicAdd(ao + j * HEAD_DIM, hs[j * HEAD_DIM] * al[j]);
}

// ---------------------------------------------------------------------------
// out = relu(acc + bias)   (in-place safe)
// ---------------------------------------------------------------------------
__global__ void gat_bias_relu(const float* __restrict__ acc,
                              const float* __restrict__ bias,
                              float* __restrict__ out, long long n)
{
    const long long i = (long long)blockIdx.x * blockDim.x + threadIdx.x;
    if (i >= n) return;
    const float v = acc[i] + bias[i & (HIDDEN - 1)];
    out[i] = fmaxf(v, 0.f);
}

// ---------------------------------------------------------------------------
extern "C" void kernel_launch(void* const* d_in, const int* in_sizes, int n_in,
                              void* d_out, int out_size, void* d_ws, size_t ws_size,
                              hipStream_t stream)
{
    const float* x   = (const float*)d_in[0];
    const int*   ei  = (const int*)d_in[1];
    const float* W1  = (const float*)d_in[2];
    const float* as1 = (const float*)d_in[3];
    const float* ad1 = (const float*)d_in[4];
    const float* b1  = (const float*)d_in[5];
    const float* W2  = (const float*)d_in[6];
    const float* as2 = (const float*)d_in[7];
    const float* ad2 = (const float*)d_in[8];
    const float* b2  = (const float*)d_in[9];

    const int N = in_sizes[0] / HIDDEN;
    const int E = in_sizes[1] / 2;
    const long long EE = (long long)E + N;

    float* ws = (float*)d_ws;
    float*    H   = ws;               ws += (long long)N * HIDDEN;
    float*    X2  = ws;               ws += (long long)N * HIDDEN;
    float*    AS  = ws;               ws += (long long)N * HEADS;
    float*    AD  = ws;               ws += (long long)N * HEADS;
    unsigned* MK  = (unsigned*)ws;    ws += (long long)N * HEADS;
    float*    DEN = ws;               ws += (long long)N * HEADS;
    float*    EX  = ws;

    const long long nNodeFeat = (long long)N * HIDDEN;
    const long long nNodeHead = (long long)N * HEADS;
    const long long nEdgeHead = EE * HEADS;

    auto blocks = [](long long n, int b) { return (unsigned)((n + b - 1) / b); };

    auto layer = [&](const float* in, const float* W, const float* as,
                     const float* ad, const float* bias, float* acc, float* out) {
        // init scratch for this layer (deterministic every call)
        gat_fill_u32<<<blocks(nNodeFeat, 256), 256, 0, stream>>>((unsigned*)acc, 0u, nNodeFeat);
        gat_fill_u32<<<blocks(nNodeHead, 256), 256, 0, stream>>>(MK, 0u, nNodeHead);
        gat_fill_u32<<<blocks(nNodeHead, 256), 256, 0, stream>>>((unsigned*)DEN, 0u, nNodeHead);
        // dense projection (WMMA fp32)
        gat_gemm_wmma<<<blocks(N, 16), 256, 0, stream>>>(in, W, H, N);
        // attention coefficients
        gat_attn_coeff<<<blocks(nNodeHead, 256), 256, 0, stream>>>(H, as, ad, AS, AD, N);
        // edge softmax
        gat_edge_max<<<blocks(nEdgeHead, 256), 256, 0, stream>>>(AS, AD, ei, E, EE, MK);
        gat_edge_exp<<<blocks(nEdgeHead, 256), 256, 0, stream>>>(AS, AD, ei, E, EE, MK, EX, DEN);
        // weighted aggregation (one wave per edge)
        gat_edge_aggr<<<blocks(EE, 8), 256, 0, stream>>>(H, EX, DEN, ei, E, EE, acc);
        // bias + relu
        gat_bias_relu<<<blocks(nNodeFeat, 256), 256, 0, stream>>>(acc, bias, out, nNodeFeat);
    };

    float* out = (float*)d_out;
    layer(x,  W1, as1, ad1, b1, X2,  X2);   // layer 1 -> X2
    layer(X2, W2, as2, ad2, b2, out, out);  // layer 2 -> d_out
}